// TemporalSelfAttention_56435870269919
// MI455X (gfx1250) — compile-verified
//
#include <hip/hip_runtime.h>

typedef float v2f __attribute__((ext_vector_type(2)));
typedef float v8f __attribute__((ext_vector_type(8)));
typedef int   v4i __attribute__((ext_vector_type(4)));

typedef __attribute__((address_space(1))) v4i gv4i;  // global int4
typedef __attribute__((address_space(3))) v4i lv4i;  // LDS int4

#define S_LEN   1024
#define EMBD    256
#define NH      8
#define HD      32
#define KPAD    36                   // padded K row stride (floats): conflict-free, 16B aligned
#define PPAD    17                   // padded P row stride (floats)
#define WAVES   4
#define QTILE   16
#define QBLK    (WAVES * QTILE)      // 64 queries per workgroup
#define KSTAGE  128                  // keys staged in LDS per iteration
#define NSTAGE  (S_LEN / KSTAGE)
#define KBLKS   (KSTAGE / 16)
#define AOPS    8                    // async b128 ops per thread per stage

// Fused DPP16 ROW_XMASK butterfly steps: one VOP2+DPP instruction per step.
#define DPP_MAX_STEP(x, m)                                                   \
  asm("v_max_num_f32_dpp %0, %0, %0 row_xmask:" #m                           \
      " row_mask:0xf bank_mask:0xf bound_ctrl:1" : "+v"(x))
#define DPP_ADD_STEP(x, m)                                                   \
  asm("v_add_f32_dpp %0, %0, %0 row_xmask:" #m                               \
      " row_mask:0xf bank_mask:0xf bound_ctrl:1" : "+v"(x))

static __device__ __forceinline__ float row_max16(float x) {
  DPP_MAX_STEP(x, 1);
  DPP_MAX_STEP(x, 2);
  DPP_MAX_STEP(x, 4);
  DPP_MAX_STEP(x, 8);
  return x;
}

static __device__ __forceinline__ float row_sum16(float x) {
  DPP_ADD_STEP(x, 1);
  DPP_ADD_STEP(x, 2);
  DPP_ADD_STEP(x, 4);
  DPP_ADD_STEP(x, 8);
  return x;
}

// Compiler waitcnt insertion doesn't track ASYNCcnt for our ops: wait manually.
#define WAIT_ASYNC(n) asm volatile("s_wait_asynccnt " #n ::: "memory")

__launch_bounds__(WAVES * 32)
__global__ void tsa_attn_kernel(const float* __restrict__ hist,
                                const float* __restrict__ query,
                                float* __restrict__ out) {
  __shared__ float lds_k[2][KSTAGE * KPAD];       // 2 x 18 KB: double-buffered K (=V)
  __shared__ float lds_p[WAVES][QTILE * PPAD];    // ~4.25 KB: per-wave P transpose

  const int h    = blockIdx.y;
  const int qb   = blockIdx.x;
  const int tid  = threadIdx.x;
  const int wave = tid >> 5;
  const int lane = tid & 31;
  const int lo   = lane & 15;    // lane id within 16-lane half
  const int hi   = lane >> 4;    // which half

  const int qrow0 = qb * QBLK + wave * QTILE;

  const float scale = 0.17677669529663687f;  // 1/sqrt(32), folded into Q

  // Async-copy one K stage (KSTAGE x 32 floats) into an LDS buffer:
  // 128 threads x 8 x b128 = 16 KB, memory -> LDS directly (no VGPR round trip).
  auto issue_stage = [&](int s, float* dstbuf) {
    const float4* kg  = (const float4*)(hist + (size_t)(s * KSTAGE) * EMBD + h * HD);
    float4*       lk4 = (float4*)dstbuf;
#pragma unroll
    for (int i = 0; i < AOPS; ++i) {
      int linear = tid + i * (WAVES * 32);   // 0..1023 float4 slots
      int key = linear >> 3;
      int d4  = linear & 7;
      __builtin_amdgcn_global_load_async_to_lds_b128(
          (gv4i*)(kg + key * (EMBD / 4) + d4),
          (lv4i*)(lk4 + key * (KPAD / 4) + d4), 0, 0);
    }
  };

  // ---- Load Q A-fragments: A[m][k] -> lane m+16*(k>>1), vgpr k&1.
  float qa[16];
  {
    const float* qbase = query + (size_t)(qrow0 + lo) * EMBD + h * HD + 2 * hi;
#pragma unroll
    for (int c = 0; c < 8; ++c) {
      float2 t = *(const float2*)(qbase + 4 * c);
      qa[2 * c]     = t.x * scale;
      qa[2 * c + 1] = t.y * scale;
    }
  }

  v8f o0 = {};   // output columns d = 0..15   (C layout)
  v8f o1 = {};   // output columns d = 16..31
  float mrow[8], lrow[8];
#pragma unroll
  for (int v = 0; v < 8; ++v) { mrow[v] = -1e30f; lrow[v] = 0.0f; }

  issue_stage(0, lds_k[0]);   // prologue: fill buffer 0

  for (int s = 0; s < NSTAGE; ++s) {
    const int cur = s & 1;
    // Refill the other buffer for stage s+1; safe: end-of-iteration barrier
    // guarantees everyone finished reading it (stage s-1).
    if (s + 1 < NSTAGE) {
      issue_stage(s + 1, lds_k[cur ^ 1]);
      WAIT_ASYNC(8);   // in-order completion: oldest 8 (current stage) done
    } else {
      WAIT_ASYNC(0);
    }
    __syncthreads();   // publish all waves' portions of the current buffer

    for (int kb = 0; kb < KBLKS; ++kb) {
      const float* kblk = lds_k[cur] + kb * 16 * KPAD;

      // ---- scores S(16q x 16k) = Q(16x32) * K(16x32)^T, fp32 WMMA, K chunks of 4
      v8f sc = {};
#pragma unroll
      for (int c = 0; c < 8; ++c) {
        v2f a; a.x = qa[2 * c]; a.y = qa[2 * c + 1];
        // B[dd][n_k]: lane = n + 16*(dd>>1), vgpr = dd&1 -> dd = 2*hi + j, n = lo
        v2f b;
        b.x = kblk[lo * KPAD + 4 * c + 2 * hi + 0];
        b.y = kblk[lo * KPAD + 4 * c + 2 * hi + 1];
        sc = __builtin_amdgcn_wmma_f32_16x16x4_f32(false, a, false, b,
                                                   (short)0, sc, false, false);
      }

      // ---- online softmax (row m = v + 8*hi spans the 16-lane half, n = lo)
      float p[8];
#pragma unroll
      for (int v = 0; v < 8; ++v) {
        float x    = sc[v];
        float rmax = row_max16(x);
        float newm = fmaxf(mrow[v], rmax);
        float pe   = __expf(x - newm);
        float corr = __expf(mrow[v] - newm);
        float rsum = row_sum16(pe);
        lrow[v] = lrow[v] * corr + rsum;
        mrow[v] = newm;
        o0[v] *= corr;
        o1[v] *= corr;
        p[v] = pe;
      }

      // ---- transpose P (C layout -> A layout) via wave-private LDS tile.
      // Same-wave DS ops are in-order; no workgroup barrier needed.
#pragma unroll
      for (int v = 0; v < 8; ++v)
        lds_p[wave][(v + 8 * hi) * PPAD + lo] = p[v];
      asm volatile("" ::: "memory");

      float pa[8];
#pragma unroll
      for (int c2 = 0; c2 < 4; ++c2) {
        // A[m][k]: m = lo, k = 4*c2 + 2*hi + j
        pa[2 * c2]     = lds_p[wave][lo * PPAD + 4 * c2 + 2 * hi + 0];
        pa[2 * c2 + 1] = lds_p[wave][lo * PPAD + 4 * c2 + 2 * hi + 1];
      }

      // ---- O += P(16x16) * V(16x32), 4-key chunks, two 16-wide d halves
#pragma unroll
      for (int c2 = 0; c2 < 4; ++c2) {
        v2f a; a.x = pa[2 * c2]; a.y = pa[2 * c2 + 1];
        // B[kk][n_d]: kk = 2*hi + j -> V row = 4*c2 + 2*hi + j, col = e*16 + lo
        int r0 = (4 * c2 + 2 * hi) * KPAD;
        v2f b0, b1;
        b0.x = kblk[r0 + lo];
        b0.y = kblk[r0 + KPAD + lo];
        o0 = __builtin_amdgcn_wmma_f32_16x16x4_f32(false, a, false, b0,
                                                   (short)0, o0, false, false);
        b1.x = kblk[r0 + 16 + lo];
        b1.y = kblk[r0 + KPAD + 16 + lo];
        o1 = __builtin_amdgcn_wmma_f32_16x16x4_f32(false, a, false, b1,
                                                   (short)0, o1, false, false);
      }
    }
    __syncthreads();   // all reads of lds_k[cur] done before it is refilled
  }

  // ---- epilogue: normalize and store (out[s][h*32 + d])
#pragma unroll
  for (int v = 0; v < 8; ++v) {
    float inv = 1.0f / lrow[v];
    size_t row = (size_t)(qrow0 + v + 8 * hi) * EMBD + h * HD;
    out[row + lo]      = o0[v] * inv;
    out[row + 16 + lo] = o1[v] * inv;
  }
}

extern "C" void kernel_launch(void* const* d_in, const int* in_sizes, int n_in,
                              void* d_out, int out_size, void* d_ws, size_t ws_size,
                              hipStream_t stream) {
  const float* hist  = (const float*)d_in[0];   // history_bev (1,1024,256) fp32
  const float* query = (const float*)d_in[1];   // bev_query   (1,1024,256) fp32
  float* out = (float*)d_out;                   // (1,1024,256) fp32

  dim3 grid(S_LEN / QBLK, NH);   // (16, 8)
  dim3 block(WAVES * 32);        // 128 threads = 4 waves
  tsa_attn_kernel<<<grid, block, 0, stream>>>(hist, query, out);
}